// CircularNN_58660663329370
// MI455X (gfx1250) — compile-verified
//
#include <hip/hip_runtime.h>
#include <math.h>

#define NFEAT   784
#define ROWS    16            // rows/block for unfused layers
#define ROWS_F  8             // rows/block for fused L1+L2 (two LDS tiles)
#define THREADS 256
#define CHUNKS  (NFEAT / 4)   // 196 16-byte chunks per row

// ---- CDNA5 async tensor-data path: global -> LDS DMA, tracked by ASYNCcnt ----
__device__ __forceinline__ void async_load16(unsigned lds_off, const float* gaddr) {
    // VDST = LDS byte address (low 32 bits of the generic LDS pointer),
    // VADDR = 64-bit global address, no SADDR.
    asm volatile("global_load_async_to_lds_b128 %0, %1, off"
                 :: "v"(lds_off), "v"(gaddr)
                 : "memory");
}
__device__ __forceinline__ void wait_async0() {
    asm volatile("s_wait_asynccnt 0x0" ::: "memory");
}

__device__ __forceinline__ float gelu_exact(float x) {
    return 0.5f * x * (1.0f + erff(x * 0.70710678118654752f));
}

// ---------------------------------------------------------------------------
// Fused L1+L2: x --async--> LDS, h1 = gelu(sp1) kept entirely in LDS,
// t2 = gelu(sp2(h1)) -> global, plus per-block BN partials. h1 never hits HBM.
// ---------------------------------------------------------------------------
__global__ __launch_bounds__(THREADS)
void fused12_kernel(const float* __restrict__ x,     // [B, N]
                    const int*   __restrict__ idx1,  // [N, 2]
                    const float* __restrict__ W1,    // [N, 2]
                    const float* __restrict__ b1,    // [N]
                    const int*   __restrict__ idx2,  // [N, 4]
                    const float* __restrict__ W2,    // [N, 4]
                    const float* __restrict__ b2,    // [N]
                    float*       __restrict__ out,   // [B, N]  (t2)
                    float*       __restrict__ part_sum, // [nblocks, N]
                    float*       __restrict__ part_sq)  // [nblocks, N]
{
    __shared__ __align__(16) float smem_x[ROWS_F * NFEAT];
    __shared__ __align__(16) float smem_h[ROWS_F * NFEAT];

    const int tid = threadIdx.x;
    const long long row0 = (long long)blockIdx.x * ROWS_F;
    const float* base = x + row0 * NFEAT;

    // Stage 8 rows of x asynchronously.
    for (int i = tid; i < ROWS_F * CHUNKS; i += THREADS) {
        unsigned lds_off = (unsigned)(uintptr_t)(const void*)&smem_x[i * 4];
        async_load16(lds_off, base + i * 4);
    }
    wait_async0();
    __syncthreads();

    // Layer 1: h1 = gelu(sp1(x)) -> LDS only.
    for (int j = tid; j < NFEAT; j += THREADS) {
        const int   i0 = idx1[j * 2 + 0], i1 = idx1[j * 2 + 1];
        const float w0 = W1[j * 2 + 0],  w1 = W1[j * 2 + 1];
        const float jb = b1[j];
        for (int r = 0; r < ROWS_F; ++r) {
            const float* srow = smem_x + r * NFEAT;
            float acc = fmaf(srow[i1], w1, fmaf(srow[i0], w0, jb));
            smem_h[r * NFEAT + j] = gelu_exact(acc);
        }
    }
    __syncthreads();

    // Layer 2: t2 = gelu(sp2(h1)) -> global, with BN partials.
    for (int j = tid; j < NFEAT; j += THREADS) {
        int   ji[4];
        float jw[4];
#pragma unroll
        for (int k = 0; k < 4; ++k) {
            ji[k] = idx2[j * 4 + k];
            jw[k] = W2[j * 4 + k];
        }
        const float jb = b2[j];
        float psum = 0.0f, psq = 0.0f;
        for (int r = 0; r < ROWS_F; ++r) {
            const float* srow = smem_h + r * NFEAT;
            float acc = jb;
#pragma unroll
            for (int k = 0; k < 4; ++k)
                acc = fmaf(srow[ji[k]], jw[k], acc);
            float act = gelu_exact(acc);
            out[(row0 + r) * NFEAT + j] = act;   // coalesced across lanes
            psum += act;
            psq = fmaf(act, act, psq);
        }
        part_sum[(size_t)blockIdx.x * NFEAT + j] = psum;
        part_sq [(size_t)blockIdx.x * NFEAT + j] = psq;
    }
}

// ---------------------------------------------------------------------------
// Generic fused layer: async-stage 16 rows, apply folded-BN affine in LDS
// (vectorized b128), sparse-gather + activation, optional BN partials.
// ---------------------------------------------------------------------------
template <int K, bool AFFINE, bool STATS, bool RELU_ACT>
__global__ __launch_bounds__(THREADS)
void layer_kernel(const float* __restrict__ in,       // [B, N]
                  const float* __restrict__ aff_a,    // [N] or null
                  const float* __restrict__ aff_d,    // [N] or null
                  const int*   __restrict__ idx,      // [N, K]
                  const float* __restrict__ W,        // [N, K]
                  const float* __restrict__ bias,     // [N]
                  float*       __restrict__ out,      // [B, N]
                  float*       __restrict__ part_sum, // [nblocks, N] or null
                  float*       __restrict__ part_sq)  // [nblocks, N] or null
{
    __shared__ __align__(16) float smem[ROWS * NFEAT];
    __shared__ __align__(16) float s_a[NFEAT];
    __shared__ __align__(16) float s_d[NFEAT];

    const int tid = threadIdx.x;
    const long long row0 = (long long)blockIdx.x * ROWS;
    const float* base = in + row0 * NFEAT;

    for (int i = tid; i < ROWS * CHUNKS; i += THREADS) {
        unsigned lds_off = (unsigned)(uintptr_t)(const void*)&smem[i * 4];
        async_load16(lds_off, base + i * 4);
    }
    if (AFFINE) {
        for (int c = tid; c < NFEAT; c += THREADS) {
            s_a[c] = aff_a[c];
            s_d[c] = aff_d[c];
        }
    }
    wait_async0();
    __syncthreads();

    if (AFFINE) {
        float4* sm4 = reinterpret_cast<float4*>(smem);
        const float4* a4p = reinterpret_cast<const float4*>(s_a);
        const float4* d4p = reinterpret_cast<const float4*>(s_d);
        for (int c4 = tid; c4 < NFEAT / 4; c4 += THREADS) {
            const float4 av = a4p[c4];
            const float4 dv = d4p[c4];
#pragma unroll
            for (int r = 0; r < ROWS; ++r) {
                float4 v = sm4[r * (NFEAT / 4) + c4];
                v.x = fmaf(av.x, v.x, dv.x);
                v.y = fmaf(av.y, v.y, dv.y);
                v.z = fmaf(av.z, v.z, dv.z);
                v.w = fmaf(av.w, v.w, dv.w);
                sm4[r * (NFEAT / 4) + c4] = v;
            }
        }
        __syncthreads();
    }

    for (int j = tid; j < NFEAT; j += THREADS) {
        int   ji[K];
        float jw[K];
#pragma unroll
        for (int k = 0; k < K; ++k) {
            ji[k] = idx[j * K + k];
            jw[k] = W[j * K + k];
        }
        const float jb = bias[j];
        float psum = 0.0f, psq = 0.0f;
        for (int r = 0; r < ROWS; ++r) {
            const float* srow = smem + r * NFEAT;
            float acc = jb;
#pragma unroll
            for (int k = 0; k < K; ++k)
                acc = fmaf(srow[ji[k]], jw[k], acc);
            float act = RELU_ACT ? fmaxf(acc, 0.0f) : gelu_exact(acc);
            out[(row0 + r) * NFEAT + j] = act;
            if (STATS) { psum += act; psq = fmaf(act, act, psq); }
        }
        if (STATS) {
            part_sum[(size_t)blockIdx.x * NFEAT + j] = psum;
            part_sq [(size_t)blockIdx.x * NFEAT + j] = psq;
        }
    }
}

// One block per column: tree-reduce per-block partials, emit folded affine
// a = g*rsqrt(var+eps), d = be - a*mean.
__global__ __launch_bounds__(THREADS)
void bn_reduce_kernel(const float* __restrict__ part_sum, // [nblocks, N]
                      const float* __restrict__ part_sq,  // [nblocks, N]
                      const float* __restrict__ gamma,
                      const float* __restrict__ beta,
                      float* __restrict__ a,
                      float* __restrict__ d,
                      int nblocks, float invB)
{
    __shared__ float red_s[THREADS];
    __shared__ float red_q[THREADS];
    const int j = blockIdx.x;
    const int t = threadIdx.x;
    float s = 0.0f, q = 0.0f;
    for (int b = t; b < nblocks; b += THREADS) {
        s += part_sum[(size_t)b * NFEAT + j];
        q += part_sq [(size_t)b * NFEAT + j];
    }
    red_s[t] = s;
    red_q[t] = q;
    __syncthreads();
#pragma unroll
    for (int off = THREADS / 2; off > 0; off >>= 1) {
        if (t < off) {
            red_s[t] += red_s[t + off];
            red_q[t] += red_q[t + off];
        }
        __syncthreads();
    }
    if (t == 0) {
        float m  = red_s[0] * invB;
        float v  = fmaf(-m, m, red_q[0] * invB);
        float sc = gamma[j] * rsqrtf(v + 1e-5f);
        a[j] = sc;
        d[j] = fmaf(-sc, m, beta[j]);
    }
}

extern "C" void kernel_launch(void* const* d_in, const int* in_sizes, int n_in,
                              void* d_out, int out_size, void* d_ws, size_t ws_size,
                              hipStream_t stream)
{
    const float* x    = (const float*)d_in[0];
    const int*   idx1 = (const int*)  d_in[1];
    const float* W1   = (const float*)d_in[2];
    const float* b1   = (const float*)d_in[3];
    const int*   idx2 = (const int*)  d_in[4];
    const float* W2   = (const float*)d_in[5];
    const float* b2   = (const float*)d_in[6];
    const int*   idx3 = (const int*)  d_in[7];
    const float* W3   = (const float*)d_in[8];
    const float* b3   = (const float*)d_in[9];
    const float* g2   = (const float*)d_in[10];
    const float* be2  = (const float*)d_in[11];
    const float* g3   = (const float*)d_in[12];
    const float* be3  = (const float*)d_in[13];

    const int N = NFEAT;
    const int B = in_sizes[0] / N;          // 16384
    const int NBLK  = B / ROWS;             // 1024 (generic layers)
    const int NBLKF = B / ROWS_F;           // 2048 (fused L1+L2)
    const float invB = 1.0f / (float)B;

    float* outb = (float*)d_out;              // ping buffer (also final output)
    float* t    = (float*)d_ws;               // pong buffer [B*N]
    float* psum = t + (size_t)B * N;          // [NBLKF, N] (max size)
    float* psq  = psum + (size_t)NBLKF * N;   // [NBLKF, N]
    float* st   = psq  + (size_t)NBLKF * N;
    float* a2 = st + 0 * N;  float* dd2 = st + 1 * N;
    float* a3 = st + 2 * N;  float* dd3 = st + 3 * N;
    float* a4 = st + 4 * N;  float* dd4 = st + 5 * N;

    dim3 blk(THREADS);
    dim3 rgrid(N);

    // L1+L2 fused: t2 = gelu(sp2(gelu(sp1(x)))) + bn2 partials   -> t
    fused12_kernel<<<dim3(NBLKF), blk, 0, stream>>>(
        x, idx1, W1, b1, idx2, W2, b2, t, psum, psq);
    bn_reduce_kernel<<<rgrid, blk, 0, stream>>>(psum, psq, g2, be2, a2, dd2, NBLKF, invB);

    // L3: t3 = gelu(sp3(bn2(t2))) + bn3 partials                 -> outb
    layer_kernel<8, true, true, false><<<dim3(NBLK), blk, 0, stream>>>(
        t, a2, dd2, idx3, W3, b3, outb, psum, psq);
    bn_reduce_kernel<<<rgrid, blk, 0, stream>>>(psum, psq, g3, be3, a3, dd3, NBLK, invB);

    // L4: t4 = gelu(sp2(bn3(t3))) + bn2' partials                -> t
    layer_kernel<4, true, true, false><<<dim3(NBLK), blk, 0, stream>>>(
        outb, a3, dd3, idx2, W2, b2, t, psum, psq);
    bn_reduce_kernel<<<rgrid, blk, 0, stream>>>(psum, psq, g2, be2, a4, dd4, NBLK, invB);

    // L5: out = relu(sp1(bn2'(t4)))                              -> outb (final)
    layer_kernel<2, true, false, true><<<dim3(NBLK), blk, 0, stream>>>(
        t, a4, dd4, idx1, W1, b1, outb, nullptr, nullptr);
}